// KldLossWithSample_54168127537391
// MI455X (gfx1250) — compile-verified
//
#include <hip/hip_runtime.h>
#include <hip/hip_bf16.h>
#include <cstdint>
#include <cstddef>

#define TRAIN_NODE_NONE 31999
#define NEG_BIG (-3.0e38f)

// ---------------------------------------------------------------------------
// gfx1250 async global->LDS path (ASYNCcnt-tracked), with safe fallback.
// Builtin prototype (from hipcc diagnostic):
//   void __builtin_amdgcn_global_load_async_to_lds_b128(
//       int4 addrspace(1)* src, int4 addrspace(3)* dst, imm int off, imm int cpol)
// ---------------------------------------------------------------------------
#if defined(__AMDGCN__) && \
    __has_builtin(__builtin_amdgcn_global_load_async_to_lds_b128) && \
    __has_builtin(__builtin_amdgcn_s_wait_asynccnt)
#define KLD_HAVE_ASYNC 1
#else
#define KLD_HAVE_ASYNC 0
#endif

#if KLD_HAVE_ASYNC
typedef int kld_v4i __attribute__((ext_vector_type(4)));
typedef __attribute__((address_space(1))) kld_v4i* kld_gptr;
typedef __attribute__((address_space(3))) kld_v4i* kld_lptr;
#endif

// ---------------------------------------------------------------------------
// Online (rescaling) log-sum-exp accumulators.
//   state (m, s)    represents  sum_j exp(v_j - m) = s
//   state (m, s, t) additionally t = sum_j exp(y_j - m) * (y_j - x_j)
// ---------------------------------------------------------------------------
__device__ __forceinline__ void upd4_x(const float4 v, float& m, float& s) {
    float c  = fmaxf(fmaxf(v.x, v.y), fmaxf(v.z, v.w));
    float nm = fmaxf(m, c);
    float a  = __expf(m - nm);
    s = fmaf(s, a, __expf(v.x - nm) + __expf(v.y - nm) +
                   __expf(v.z - nm) + __expf(v.w - nm));
    m = nm;
}

__device__ __forceinline__ void upd4_y(const float4 w, const float4 v,
                                       float& m, float& s, float& t) {
    float c  = fmaxf(fmaxf(w.x, w.y), fmaxf(w.z, w.w));
    float nm = fmaxf(m, c);
    float a  = __expf(m - nm);
    float e0 = __expf(w.x - nm), e1 = __expf(w.y - nm);
    float e2 = __expf(w.z - nm), e3 = __expf(w.w - nm);
    s = fmaf(s, a, (e0 + e1) + (e2 + e3));
    float td = fmaf(e0, w.x - v.x,
               fmaf(e1, w.y - v.y,
               fmaf(e2, w.z - v.z, e3 * (w.w - v.w))));
    t = fmaf(t, a, td);
    m = nm;
}

__device__ __forceinline__ void comb2(float& m, float& s, float mo, float so) {
    float nm = fmaxf(m, mo);
    s = fmaf(s, __expf(m - nm), so * __expf(mo - nm));
    m = nm;
}

__device__ __forceinline__ void comb3(float& m, float& s, float& t,
                                      float mo, float so, float to) {
    float nm = fmaxf(m, mo);
    float a  = __expf(m - nm);
    float b  = __expf(mo - nm);
    s = fmaf(s, a, so * b);
    t = fmaf(t, a, to * b);
    m = nm;
}

// ---------------------------------------------------------------------------
// Kernel 1: one block (256 thr = 8 wave32) per row.
// Single streaming pass over x-row and y-row producing:
//   ws[row]     = kld_row
//   ws[N+row]   = softmax prob at (clamped) label
// ---------------------------------------------------------------------------
__global__ __launch_bounds__(256) void kld_row_stats(
    const float* __restrict__ x, const float* __restrict__ y,
    const int* __restrict__ labels, const int* __restrict__ nmod_p,
    float* __restrict__ ws, int V, int N)
{
    const int row = blockIdx.x;
    const int tid = threadIdx.x;
    const float* __restrict__ xr = x + (size_t)row * V;
    const float* __restrict__ yr = y + (size_t)row * V;

    // Label gather (index = max(target, 0)); overlapped with the main stream.
    float xt = 0.f;
    if (tid == 0) {
        const int nmod = nmod_p[0];
        const int lab  = labels[(size_t)row * nmod];
        const int gi   = (lab < 0 || lab == TRAIN_NODE_NONE) ? 0 : lab;
        xt = xr[gi];
    }

    float mx = NEG_BIG, sx = 0.f;
    float my = NEG_BIG, sy = 0.f, tt = 0.f;

    const int quads = V >> 2;            // float4 elements per row
    const int steps = quads / 256;       // full 256-quad tiles

    const float4* __restrict__ xq4 = (const float4*)xr;
    const float4* __restrict__ yq4 = (const float4*)yr;

#if KLD_HAVE_ASYNC
    // Double-buffered async global->LDS pipeline. Each lane owns its 16B slot,
    // so consumption needs only the per-wave ASYNCcnt wait (no barriers).
    __shared__ float4 lx[2][256];
    __shared__ float4 ly[2][256];
    if (steps > 0) {
        __builtin_amdgcn_global_load_async_to_lds_b128(
            (kld_gptr)(xq4 + tid), (kld_lptr)&lx[0][tid], 0, 0);
        __builtin_amdgcn_global_load_async_to_lds_b128(
            (kld_gptr)(yq4 + tid), (kld_lptr)&ly[0][tid], 0, 0);
    }
    for (int k = 0; k < steps; ++k) {
        const int b = k & 1;
        if (k + 1 < steps) {
            const int q = (k + 1) * 256 + tid;
            __builtin_amdgcn_global_load_async_to_lds_b128(
                (kld_gptr)(xq4 + q), (kld_lptr)&lx[b ^ 1][tid], 0, 0);
            __builtin_amdgcn_global_load_async_to_lds_b128(
                (kld_gptr)(yq4 + q), (kld_lptr)&ly[b ^ 1][tid], 0, 0);
            __builtin_amdgcn_s_wait_asynccnt(2);   // tile k landed; k+1 in flight
        } else {
            __builtin_amdgcn_s_wait_asynccnt(0);
        }
        asm volatile("" ::: "memory");
        const float4 xv = lx[b][tid];
        const float4 yv = ly[b][tid];
        upd4_x(xv, mx, sx);
        upd4_y(yv, xv, my, sy, tt);
    }
#else
    for (int k = 0; k < steps; ++k) {
        const int q = k * 256 + tid;
        const float4 xv = xq4[q];
        const float4 yv = yq4[q];
        upd4_x(xv, mx, sx);
        upd4_y(yv, xv, my, sy, tt);
    }
#endif

    // Remainder quads (V=32000 -> 64 quads; coalesced direct loads).
    for (int q = steps * 256 + tid; q < quads; q += 256) {
        const float4 xv = xq4[q];
        const float4 yv = yq4[q];
        upd4_x(xv, mx, sx);
        upd4_y(yv, xv, my, sy, tt);
    }
    // Scalar tail if V % 4 != 0 (single element == state (v, 1)).
    for (int j = (quads << 2) + tid; j < V; j += 256) {
        const float xv = xr[j], yv = yr[j];
        comb2(mx, sx, xv, 1.f);
        comb3(my, sy, tt, yv, 1.f, yv - xv);
    }

    // wave32 butterfly reduction (5 steps — gfx1250 is wave32).
    #pragma unroll
    for (int off = 16; off > 0; off >>= 1) {
        const float mo = __shfl_xor(mx, off);
        const float so = __shfl_xor(sx, off);
        const float m2 = __shfl_xor(my, off);
        const float s2 = __shfl_xor(sy, off);
        const float t2 = __shfl_xor(tt, off);
        comb2(mx, sx, mo, so);
        comb3(my, sy, tt, m2, s2, t2);
    }

    // Cross-wave combine (8 waves).
    __shared__ float red[5][8];
    const int wave = tid >> 5;
    if ((tid & 31) == 0) {
        red[0][wave] = mx; red[1][wave] = sx;
        red[2][wave] = my; red[3][wave] = sy; red[4][wave] = tt;
    }
    __syncthreads();
    if (tid == 0) {
        #pragma unroll
        for (int w = 1; w < 8; ++w) {
            comb2(mx, sx, red[0][w], red[1][w]);
            comb3(my, sy, tt, red[2][w], red[3][w], red[4][w]);
        }
        const float logZx = mx + __logf(sx);
        const float logZy = my + __logf(sy);
        ws[row]     = tt / sy - logZy + logZx;   // KL(y1 || softmax(x)) per row
        ws[N + row] = __expf(xt - logZx);        // prob at label
    }
}

// ---------------------------------------------------------------------------
// Kernel 2: mask + final masked mean over N rows (single block).
// UNIT_LIST = {5,17,42,100,256,512,1024,2048}
// ---------------------------------------------------------------------------
__global__ __launch_bounds__(256) void kld_finalize(
    const float* __restrict__ ws, const int* __restrict__ labels,
    const int* __restrict__ nmod_p, const float* __restrict__ score_p,
    float* __restrict__ out, int N)
{
    const int tid  = threadIdx.x;
    const int nmod = nmod_p[0];
    const float thr = score_p[0];

    float sl = 0.f, sm = 0.f;
    for (int i = tid; i < N; i += 256) {
        const float kld  = ws[i];
        const float prob = ws[N + i];
        const int lab = labels[(size_t)i * nmod];
        const int tgt = (lab < 0 || lab == TRAIN_NODE_NONE) ? -1 : lab;
        const bool inlist = (tgt == 5) || (tgt == 17) || (tgt == 42) ||
                            (tgt == 100) || (tgt == 256) || (tgt == 512) ||
                            (tgt == 1024) || (tgt == 2048);
        const float mask = (inlist && (prob > thr)) ? 0.f : 1.f;
        sl = fmaf(kld, mask, sl);
        sm += mask;
    }

    #pragma unroll
    for (int off = 16; off > 0; off >>= 1) {
        sl += __shfl_xor(sl, off);
        sm += __shfl_xor(sm, off);
    }

    __shared__ float rl[8], rm[8];
    const int wave = tid >> 5;
    if ((tid & 31) == 0) { rl[wave] = sl; rm[wave] = sm; }
    __syncthreads();
    if (tid == 0) {
        float L = 0.f, M = 0.f;
        #pragma unroll
        for (int w = 0; w < 8; ++w) { L += rl[w]; M += rm[w]; }
        out[0] = L / M;
    }
}

// ---------------------------------------------------------------------------
// Host launcher. Inputs (setup_inputs order):
//   0: x [B,T,V] f32   1: y [B,T,V] f32   2: frames_label [B,T,NMOD] i32
//   3: nmod (i32 scalar)   4: sample_score (f32 scalar)
// Output: 1 float. Workspace: 2*N floats.
// ---------------------------------------------------------------------------
extern "C" void kernel_launch(void* const* d_in, const int* in_sizes, int n_in,
                              void* d_out, int out_size, void* d_ws, size_t ws_size,
                              hipStream_t stream) {
    const float* x      = (const float*)d_in[0];
    const float* y      = (const float*)d_in[1];
    const int*   labels = (const int*)d_in[2];
    const int*   nmod_p = (const int*)d_in[3];
    const float* ss_p   = (const float*)d_in[4];

    // Reference fixes NMOD = 3; derive N and V from the flat sizes.
    const int NMOD = 3;
    const int N = in_sizes[2] / NMOD;          // 2048 rows
    const int V = in_sizes[0] / N;             // 32000 classes

    float* ws = (float*)d_ws;

    kld_row_stats<<<N, 256, 0, stream>>>(x, y, labels, nmod_p, ws, V, N);
    kld_finalize<<<1, 256, 0, stream>>>(ws, labels, nmod_p, ss_p, (float*)d_out, N);
}